// EEGPreprocessing_74483322847309
// MI455X (gfx1250) — compile-verified
//
#include <hip/hip_runtime.h>
#include <cstdint>

typedef float v2f __attribute__((ext_vector_type(2)));
typedef float v8f __attribute__((ext_vector_type(8)));

#define SECTIONS 10
#define BK 16            // time-block (WMMA tile M)
#define KA 20            // augmented K: 16 inputs + state z0,z1 + 2 zero pad
#define BATCH 32
#define LEN 20000
#define CH 64
#define NWAVE 4          // waves per block: one 16-channel group each
#define NBLK (LEN / BK)  // 1250

// --- CDNA5 helpers -------------------------------------------------------

// cross-half swap (SWAPX16): group-of-32 swizzle, xor_mask=0x10, and_mask=0x1f
__device__ __forceinline__ float swap16(float x) {
  return __int_as_float(__builtin_amdgcn_ds_swizzle(__float_as_int(x), 0x401F));
}

// async global <-> LDS copies, 16B per lane (tracked by ASYNCcnt)
__device__ __forceinline__ void async_load_b128(uint32_t lds_addr, uint64_t gaddr) {
  asm volatile("global_load_async_to_lds_b128 %0, %1, off"
               :: "v"(lds_addr), "v"(gaddr) : "memory");
}
__device__ __forceinline__ void async_store_b128(uint64_t gaddr, uint32_t lds_addr) {
  asm volatile("global_store_async_from_lds_b128 %0, %1, off"
               :: "v"(gaddr), "v"(lds_addr) : "memory");
}
__device__ __forceinline__ void wait_async0() {
  asm volatile("s_wait_asynccnt 0" ::: "memory");
}
__device__ __forceinline__ void wait_ds0() {
  asm volatile("s_wait_dscnt 0" ::: "memory");
}

// --- main kernel ---------------------------------------------------------
// grid = 32 (one block per batch), block = 128 (4 waves; wave w owns channels
// [16w,16w+16) of its batch). Exact block-state-space sosfilt:
//   per section, per 16-sample block:  [Y ; z'] = [T|H ; Phi|A^16] @ [X ; z]
// both products on v_wmma_f32_16x16x4_f32 with K augmented to 20.

__global__ __launch_bounds__(128, 1)
void sosfilt_wmma(const float* __restrict__ x, const float* __restrict__ sos,
                  float* __restrict__ out) {
  __shared__ float sTa[SECTIONS][16][KA]; // [T (16x16) | H (16x2) | 0 0]
  __shared__ float sG [SECTIONS][16][KA]; // row0=[Phi0|A16 r0|0 0], row8=[Phi1|A16 r1|0 0]
  __shared__ float xt [2][BK][CH];        // double-buffered input tile
  __shared__ float yt [NWAVE][BK][16];    // wave-private section handoff tile
  __shared__ float yo [BK][CH];           // output staging tile (async store)

  const int tid  = threadIdx.x;
  const int b    = blockIdx.x;
  const int lane = tid & 31;
  const int wv   = tid >> 5;        // channel group 0..3
  const int hi   = lane >> 4;       // lane half
  const int chl  = lane & 15;       // channel-in-group / A-matrix row (M)
  const int tb   = 2 * hi;          // fragment K base within each K=4 chunk

  const float* xg0 = x + (size_t)b * LEN * CH;

  // kick off async load of tile 0 (4 KB contiguous; 32 B per thread)
  {
    uint64_t g = (uint64_t)(uintptr_t)xg0 + (uint64_t)tid * 32;
    uint32_t l = (uint32_t)(uintptr_t)(&xt[0][0][0]) + (uint32_t)tid * 32;
    async_load_b128(l, g);
    async_load_b128(l + 16, g + 16);
  }

  // build per-section block matrices from sos (exact linear algebra)
  if (tid < SECTIONS) {
    const int s = tid;
    float b0 = sos[s*6+0], b1 = sos[s*6+1], b2 = sos[s*6+2];
    float a1 = sos[s*6+4], a2 = sos[s*6+5];
    // DF2T:  z' = A z + bv u,  y = z0 + b0 u ; A=[[-a1,1],[-a2,0]]
    float A00 = -a1, A01 = 1.f, A10 = -a2, A11 = 0.f;
    float q0[16], q1[16];                 // q_t = A^t bv
    q0[0] = b1 - a1 * b0;
    q1[0] = b2 - a2 * b0;
    for (int t = 1; t < 16; ++t) {
      q0[t] = A00 * q0[t-1] + A01 * q1[t-1];
      q1[t] = A10 * q0[t-1] + A11 * q1[t-1];
    }
    float h[16];                          // impulse response: h0=d, hk=c.A^{k-1}b
    h[0] = b0;
    for (int t = 1; t < 16; ++t) h[t] = q0[t-1];
    float p00 = 1.f, p01 = 0.f, p10 = 0.f, p11 = 1.f;   // P_t = A^t
    for (int i = 0; i < 16; ++i) {
      for (int j = 0; j < 16; ++j) sTa[s][i][j] = (i >= j) ? h[i - j] : 0.f;
      sTa[s][i][16] = p00;   // H[i][0] = (c^T A^i)[0]
      sTa[s][i][17] = p01;   // H[i][1]
      sTa[s][i][18] = 0.f;
      sTa[s][i][19] = 0.f;
      float n00 = A00*p00 + A01*p10, n01 = A00*p01 + A01*p11;
      float n10 = A10*p00 + A11*p10, n11 = A10*p01 + A11*p11;
      p00 = n00; p01 = n01; p10 = n10; p11 = n11;
    }
    // p now = A^16
    for (int i = 0; i < 16; ++i)
      for (int j = 0; j < KA; ++j) sG[s][i][j] = 0.f;
    for (int t = 0; t < 16; ++t) { sG[s][0][t] = q0[15 - t]; sG[s][8][t] = q1[15 - t]; }
    sG[s][0][16] = p00; sG[s][0][17] = p01;
    sG[s][8][16] = p10; sG[s][8][17] = p11;
  }
  __syncthreads();

  // hoist all loop-invariant A-fragments into registers (~200 VGPRs; only
  // 128 waves exist chip-wide, so 1 wave/SIMD -> huge VGPR budget)
  v2f ta[SECTIONS][5], ga[SECTIONS][5];
#pragma unroll
  for (int s = 0; s < SECTIONS; ++s)
#pragma unroll
    for (int c = 0; c < 5; ++c) {
      ta[s][c] = *(const v2f*)&sTa[s][chl][4 * c + tb];
      ga[s][c] = *(const v2f*)&sG [s][chl][4 * c + tb];
    }

  float zvec[SECTIONS];   // per-section state: lanes0-15 = z0[ch], lanes16-31 = z1[ch]
#pragma unroll
  for (int s = 0; s < SECTIONS; ++s) zvec[s] = 0.f;

  int buf = 0;
  for (int nb = 0; nb < NBLK; ++nb) {
    __syncthreads();   // (A) everyone done with tile buf^1 and with yo
    wait_async0();     // my async loads for tile buf (and my yo stores) landed
    __syncthreads();   // (B) all waves' async work done -> xt[buf], yo free
    if (nb + 1 < NBLK) {
      uint64_t g = (uint64_t)(uintptr_t)(xg0 + (size_t)(nb + 1) * BK * CH)
                 + (uint64_t)tid * 32;
      uint32_t l = (uint32_t)(uintptr_t)(&xt[buf ^ 1][0][0]) + (uint32_t)tid * 32;
      async_load_b128(l, g);
      async_load_b128(l + 16, g + 16);
    }

    const float* src = &xt[buf][0][wv * 16];
    int stride = CH;

#pragma unroll
    for (int s = 0; s < SECTIONS; ++s) {
      // ---- B fragments of augmented input [X ; z0 ; z1 ; 0 ; 0] ----
      v2f bf0, bf1, bf2, bf3, bf4;
      bf0.x = src[(tb + 0) * stride + chl]; bf0.y = src[(tb + 1) * stride + chl];
      bf1.x = src[(tb + 4) * stride + chl]; bf1.y = src[(tb + 5) * stride + chl];
      bf2.x = src[(tb + 8) * stride + chl]; bf2.y = src[(tb + 9) * stride + chl];
      bf3.x = src[(tb + 12) * stride + chl]; bf3.y = src[(tb + 13) * stride + chl];
      float zs = swap16(zvec[s]);          // lanes0-15 <- z1[ch]
      bf4.x = hi ? 0.f : zvec[s];          // K=16 row: z0 ; K=18 row: 0
      bf4.y = hi ? 0.f : zs;               // K=17 row: z1 ; K=19 row: 0
      // ---- Y = [T|H] @ [X;z]  and  z' = [Phi|A16] @ [X;z] ----
      v8f acc = {0.f, 0.f, 0.f, 0.f, 0.f, 0.f, 0.f, 0.f};
      v8f zac = {0.f, 0.f, 0.f, 0.f, 0.f, 0.f, 0.f, 0.f};
      acc = __builtin_amdgcn_wmma_f32_16x16x4_f32(false, ta[s][0], false, bf0, (short)0, acc, false, false);
      zac = __builtin_amdgcn_wmma_f32_16x16x4_f32(false, ga[s][0], false, bf0, (short)0, zac, false, false);
      acc = __builtin_amdgcn_wmma_f32_16x16x4_f32(false, ta[s][1], false, bf1, (short)0, acc, false, false);
      zac = __builtin_amdgcn_wmma_f32_16x16x4_f32(false, ga[s][1], false, bf1, (short)0, zac, false, false);
      acc = __builtin_amdgcn_wmma_f32_16x16x4_f32(false, ta[s][2], false, bf2, (short)0, acc, false, false);
      zac = __builtin_amdgcn_wmma_f32_16x16x4_f32(false, ga[s][2], false, bf2, (short)0, zac, false, false);
      acc = __builtin_amdgcn_wmma_f32_16x16x4_f32(false, ta[s][3], false, bf3, (short)0, acc, false, false);
      zac = __builtin_amdgcn_wmma_f32_16x16x4_f32(false, ga[s][3], false, bf3, (short)0, zac, false, false);
      acc = __builtin_amdgcn_wmma_f32_16x16x4_f32(false, ta[s][4], false, bf4, (short)0, acc, false, false);
      zac = __builtin_amdgcn_wmma_f32_16x16x4_f32(false, ga[s][4], false, bf4, (short)0, zac, false, false);
      // D VGPR0: lanes0-15 = row0 = z0', lanes16-31 = row8 = z1'
      zvec[s] = zac[0];

      if (s < SECTIONS - 1) {
        // hand Y to next section via wave-private LDS tile
#pragma unroll
        for (int r = 0; r < 8; ++r) yt[wv][r + 8 * hi][chl] = acc[r];
        wait_ds0();                    // DS in-order per wave; fence compiler+hw
        src = &yt[wv][0][0];
        stride = 16;
      } else {
        // stage final output tile in LDS for one contiguous async store
#pragma unroll
        for (int r = 0; r < 8; ++r) yo[r + 8 * hi][wv * 16 + chl] = acc[r];
      }
    }

    wait_ds0();
    __syncthreads();   // (C) yo complete across all waves
    {
      uint64_t g = (uint64_t)(uintptr_t)(out + ((size_t)b * LEN + (size_t)nb * BK) * CH)
                 + (uint64_t)tid * 32;
      uint32_t l = (uint32_t)(uintptr_t)(&yo[0][0]) + (uint32_t)tid * 32;
      async_store_b128(g, l);
      async_store_b128(g + 16, l + 16);
    }
    buf ^= 1;
  }
  // s_endpgm performs an implicit wait-idle, covering the final async stores
}

extern "C" void kernel_launch(void* const* d_in, const int* in_sizes, int n_in,
                              void* d_out, int out_size, void* d_ws, size_t ws_size,
                              hipStream_t stream) {
  (void)in_sizes; (void)n_in; (void)out_size; (void)d_ws; (void)ws_size;
  const float* x   = (const float*)d_in[0];
  const float* sos = (const float*)d_in[1];
  float* out = (float*)d_out;
  sosfilt_wmma<<<dim3(BATCH), dim3(NWAVE * 32), 0, stream>>>(x, sos, out);
}